// VQVAE_67723044323564
// MI455X (gfx1250) — compile-verified
//
#include <hip/hip_runtime.h>
#include <hip/hip_bf16.h>

// ---------------------------------------------------------------------------
// VQ-VAE forward for MI455X (gfx1250, wave32, WMMA).
// All GEMM-like work runs on V_WMMA_F32_16X16X4_F32 (fp32 in == fp32 math,
// preserves argmin fidelity). ~172 GFLOP fp32 vs ~0.6 GB HBM traffic ->
// matrix-ALU bound; layout chosen for operand reuse (LDS-shared B tiles,
// register-resident A tiles in the VQ search).
// ---------------------------------------------------------------------------

typedef float v2f __attribute__((ext_vector_type(2)));
typedef float v8f __attribute__((ext_vector_type(8)));

static constexpr int B_    = 32;
static constexpr int CIN_C = 128;
static constexpr int T_    = 4096;
static constexpr int HID_  = 512;
static constexpr int D_    = 128;
static constexpr int KCB_  = 8192;
static constexpr int T2_   = 2048;
static constexpr int T4_   = 1024;
static constexpr int NTOK_ = B_ * T4_;      // 32768
static constexpr int CICHUNK = 128;

// ---------------------------------------------------------------------------
// Strided conv1d (k=4, s=2, p=1) as GEMM: M=Cout tile 16, N=time tile 16,
// K = Cin*4 consumed one Cin (4 taps) per WMMA.
// Block = 128 threads (4 waves); waves share the LDS x-window, each owns a
// 16-wide Cout stripe. A-frag: lanes 0-15 row=lane, taps {0,1}; lanes 16-31
// row=lane-16, taps {2,3}.  B-frag mirrors with N instead of M.
// ---------------------------------------------------------------------------
template <int CIN, bool RELU>
__global__ __launch_bounds__(128) void conv_s2(
    const float* __restrict__ x, const float* __restrict__ w,
    const float* __restrict__ bias, float* __restrict__ y,
    int Tin, int Tout, int Cout) {
  __shared__ float xs[CICHUNK * 34];
  const int t0   = blockIdx.x * 16;
  const int b    = blockIdx.z;
  const int lane = threadIdx.x & 31;
  const int wave = threadIdx.x >> 5;
  const int co0  = blockIdx.y * 64 + wave * 16;
  const int m    = lane & 15;
  const int kh   = lane >> 4;       // 0: K%4 in {0,1}; 1: K%4 in {2,3}
  const int sbase = 2 * t0 - 1;     // left edge of window (pad=1)

  v8f acc = {};
  for (int ci0 = 0; ci0 < CIN; ci0 += CICHUNK) {
    __syncthreads();
    for (int e = threadIdx.x; e < CICHUNK * 34; e += 128) {
      int cc = e / 34, i = e % 34;
      int s = sbase + i;
      float v = 0.0f;
      if (s >= 0 && s < Tin)
        v = x[((size_t)b * CIN + (ci0 + cc)) * Tin + s];
      xs[cc * 34 + i] = v;
    }
    __syncthreads();
    const float* wrow = w + (((size_t)(co0 + m)) * CIN + ci0) * 4 + kh * 2;
#pragma unroll 4
    for (int cc = 0; cc < CICHUNK; ++cc) {
      const float2 av = *(const float2*)(wrow + (size_t)cc * 4);
      const float2 bv = *(const float2*)&xs[cc * 34 + 2 * m + 2 * kh];
      v2f a;  a.x = av.x;  a.y = av.y;
      v2f bf; bf.x = bv.x; bf.y = bv.y;
      acc = __builtin_amdgcn_wmma_f32_16x16x4_f32(false, a, false, bf,
                                                  (short)0, acc, false, false);
    }
  }
  // D layout: VGPR v, lanes 0-15 -> row v, lanes 16-31 -> row 8+v; col = lane%16
  const int t = t0 + m;
#pragma unroll
  for (int v = 0; v < 8; ++v) {
    int co = co0 + kh * 8 + v;
    float r = acc[v] + bias[co];
    if (RELU) r = fmaxf(r, 0.0f);
    y[((size_t)b * Cout + co) * Tout + t] = r;
  }
}

// ---------------------------------------------------------------------------
// ConvTranspose1d (k=4, s=2, p=1). Output parity p picks 2 taps:
//   u=2m+1: k=0 -> x[m+1], k=2 -> x[m]
//   u=2m  : k=1 -> x[m],   k=3 -> x[m-1]
// Per parity a GEMM with K = Cin*2 (two Cin per WMMA step).
// ---------------------------------------------------------------------------
template <int CIN, bool RELU>
__global__ __launch_bounds__(128) void tconv_s2(
    const float* __restrict__ x, const float* __restrict__ w,
    const float* __restrict__ bias, float* __restrict__ y,
    int Tin, int Cout) {
  __shared__ float xs[CICHUNK * 18];
  const int m0   = blockIdx.x * 16;
  const int b    = blockIdx.z;
  const int lane = threadIdx.x & 31;
  const int wave = threadIdx.x >> 5;
  const int co0  = blockIdx.y * 64 + wave * 16;
  const int m    = lane & 15;
  const int kh   = lane >> 4;
  const int Tout = Tin * 2;

  for (int par = 0; par < 2; ++par) {
    const int k0 = par ? 0 : 1, k1 = par ? 2 : 3;
    const int d0 = par ? 1 : 0, d1 = par ? 0 : -1;
    v8f acc = {};
    for (int ci0 = 0; ci0 < CIN; ci0 += CICHUNK) {
      __syncthreads();
      for (int e = threadIdx.x; e < CICHUNK * 18; e += 128) {
        int cc = e / 18, i = e % 18;          // i==0 <-> m0-1
        int s = m0 - 1 + i;
        float v = 0.0f;
        if (s >= 0 && s < Tin)
          v = x[((size_t)b * CIN + (ci0 + cc)) * Tin + s];
        xs[cc * 18 + i] = v;
      }
      __syncthreads();
#pragma unroll 4
      for (int s = 0; s < CICHUNK / 2; ++s) {
        const int cl = 2 * s + kh;            // ci within chunk for this K-half
        const int ci = ci0 + cl;
        const size_t wb = ((size_t)ci * Cout + co0 + m) * 4;
        v2f a;  a.x  = w[wb + k0];            a.y  = w[wb + k1];
        v2f bf; bf.x = xs[cl * 18 + m + 1 + d0]; bf.y = xs[cl * 18 + m + 1 + d1];
        acc = __builtin_amdgcn_wmma_f32_16x16x4_f32(false, a, false, bf,
                                                    (short)0, acc, false, false);
      }
    }
    const int u = 2 * (m0 + m) + par;
#pragma unroll
    for (int v = 0; v < 8; ++v) {
      int co = co0 + kh * 8 + v;
      float r = acc[v] + bias[co];
      if (RELU) r = fmaxf(r, 0.0f);
      y[((size_t)b * Cout + co) * Tout + u] = r;
    }
  }
}

// ---------------------------------------------------------------------------
// Codebook row squared norms.
// ---------------------------------------------------------------------------
__global__ __launch_bounds__(256) void cnorm_kernel(
    const float* __restrict__ cb, float* __restrict__ cn) {
  int j = blockIdx.x * blockDim.x + threadIdx.x;
  const float4* r = (const float4*)(cb + (size_t)j * D_);
  float s = 0.0f;
#pragma unroll
  for (int i = 0; i < D_ / 4; ++i) {
    float4 v = r[i];
    s += v.x * v.x + v.y * v.y + v.z * v.z + v.w * v.w;
  }
  cn[j] = s;
}

// ---------------------------------------------------------------------------
// VQ nearest-neighbor: score = ||c||^2 - 2*(z.c)   (||z||^2 constant per row).
// 8 waves/block; each wave holds its 16-token A-tile in registers (32 x v2f),
// codebook 16-row tile staged in LDS shared by all waves.
// NOTE: the 32-step K loop MUST be fully unrolled so af[] stays in VGPRs
// (partial unroll -> dynamic indexing -> scratch spill in the hot loop).
// ---------------------------------------------------------------------------
__global__ __launch_bounds__(256) void vq_kernel(
    const float* __restrict__ ze, const float* __restrict__ cb,
    const float* __restrict__ cnorm, int* __restrict__ idx_out,
    float* __restrict__ idxf_out) {
  __shared__ float cbs[16 * 130];   // stride 130 -> no LDS bank clustering
  __shared__ float cns[16];
  const int lane = threadIdx.x & 31;
  const int wave = threadIdx.x >> 5;
  const int n0   = blockIdx.x * 128 + wave * 16;
  const int b    = n0 >> 10;
  const int t0   = n0 & 1023;
  const int m    = lane & 15;
  const int kh   = lane >> 4;

  // A fragments: step s covers dims 4s..4s+3 (held in registers throughout)
  v2f af[32];
  const float* zb = ze + (size_t)b * D_ * T4_ + (t0 + m);
#pragma unroll
  for (int s = 0; s < 32; ++s) {
    int d = 4 * s + 2 * kh;
    af[s].x = zb[(size_t)d * T4_];
    af[s].y = zb[(size_t)(d + 1) * T4_];
  }

  float best[8];
  int   bidx[8];
#pragma unroll
  for (int v = 0; v < 8; ++v) { best[v] = 3.4e38f; bidx[v] = 0; }

  for (int j0 = 0; j0 < KCB_; j0 += 16) {
    __syncthreads();
    for (int e = threadIdx.x; e < 2048; e += 256) {
      int jl = e >> 7, d = e & 127;
      cbs[jl * 130 + d] = cb[((size_t)(j0 + jl)) * D_ + d];
    }
    if (threadIdx.x < 16) cns[threadIdx.x] = cnorm[j0 + threadIdx.x];
    if (j0 + 16 < KCB_)   // warm L2/L0 for next tile (global_prefetch_b8)
      __builtin_prefetch(&cb[((size_t)(j0 + 16)) * D_ + threadIdx.x * 8], 0, 1);
    __syncthreads();

    v8f acc = {};
#pragma unroll
    for (int s = 0; s < 32; ++s) {
      const float2 bv = *(const float2*)&cbs[m * 130 + 4 * s + 2 * kh];
      v2f bf; bf.x = bv.x; bf.y = bv.y;
      acc = __builtin_amdgcn_wmma_f32_16x16x4_f32(false, af[s], false, bf,
                                                  (short)0, acc, false, false);
    }
    const float cn = cns[m];
    const int   j  = j0 + m;
#pragma unroll
    for (int v = 0; v < 8; ++v) {
      float sc = cn - 2.0f * acc[v];
      if (sc < best[v]) { best[v] = sc; bidx[v] = j; }
    }
  }

  // min over the 16 lanes holding each C-row (first-min tie-break)
#pragma unroll
  for (int v = 0; v < 8; ++v) {
    for (int off = 8; off > 0; off >>= 1) {
      float ob = __shfl_xor(best[v], off, 16);
      int   oi = __shfl_xor(bidx[v], off, 16);
      if (ob < best[v] || (ob == best[v] && oi < bidx[v])) {
        best[v] = ob; bidx[v] = oi;
      }
    }
  }
  if (m == 0) {
#pragma unroll
    for (int v = 0; v < 8; ++v) {
      int n = n0 + kh * 8 + v;
      idx_out[n]  = bidx[v];
      idxf_out[n] = (float)bidx[v];
    }
  }
}

// ---------------------------------------------------------------------------
// z_q gather back into NCH layout.
// ---------------------------------------------------------------------------
__global__ __launch_bounds__(256) void gather_kernel(
    const float* __restrict__ cb, const int* __restrict__ idx,
    float* __restrict__ zq) {
  int g = blockIdx.x * blockDim.x + threadIdx.x;   // over NTOK_*D_
  int n = g >> 7, d = g & 127;
  int b = n >> 10, t = n & 1023;
  zq[((size_t)b * D_ + d) * T4_ + t] = cb[(size_t)idx[n] * D_ + d];
}

// ---------------------------------------------------------------------------
extern "C" void kernel_launch(void* const* d_in, const int* in_sizes, int n_in,
                              void* d_out, int out_size, void* d_ws,
                              size_t ws_size, hipStream_t stream) {
  (void)in_sizes; (void)n_in; (void)out_size; (void)ws_size;
  const float* x        = (const float*)d_in[0];
  const float* enc_w1   = (const float*)d_in[1];
  const float* enc_b1   = (const float*)d_in[2];
  const float* enc_w2   = (const float*)d_in[3];
  const float* enc_b2   = (const float*)d_in[4];
  const float* codebook = (const float*)d_in[5];
  const float* dec_w1   = (const float*)d_in[6];
  const float* dec_b1   = (const float*)d_in[7];
  const float* dec_w2   = (const float*)d_in[8];
  const float* dec_b2   = (const float*)d_in[9];

  float* out     = (float*)d_out;
  float* x_recon = out;                                  // 16777216
  float* z_q     = out + 16777216;                       //  4194304
  float* idxf    = out + 16777216 + 4194304;             //    32768
  float* z_e     = out + 16777216 + 4194304 + 32768;     //  4194304

  float* ws      = (float*)d_ws;
  float* hbuf    = ws;                                   // 33554432 (h, then g)
  float* cnorm   = ws + 33554432;                        // 8192
  int*   idx_int = (int*)(ws + 33554432 + 8192);         // 32768

  // encoder
  conv_s2<CIN_C, true><<<dim3(T2_ / 16, HID_ / 64, B_), 128, 0, stream>>>(
      x, enc_w1, enc_b1, hbuf, T_, T2_, HID_);
  conv_s2<HID_, false><<<dim3(T4_ / 16, D_ / 64, B_), 128, 0, stream>>>(
      hbuf, enc_w2, enc_b2, z_e, T2_, T4_, D_);
  // VQ
  cnorm_kernel<<<KCB_ / 256, 256, 0, stream>>>(codebook, cnorm);
  vq_kernel<<<NTOK_ / 128, 256, 0, stream>>>(z_e, codebook, cnorm, idx_int,
                                             idxf);
  gather_kernel<<<(NTOK_ * D_) / 256, 256, 0, stream>>>(codebook, idx_int,
                                                        z_q);
  // decoder (hbuf reused for g)
  tconv_s2<D_, true><<<dim3(T4_ / 16, HID_ / 64, B_), 128, 0, stream>>>(
      z_q, dec_w1, dec_b1, hbuf, T4_, HID_);
  tconv_s2<HID_, false><<<dim3(T2_ / 16, CIN_C / 64, B_), 128, 0, stream>>>(
      hbuf, dec_w2, dec_b2, x_recon, T2_, CIN_C);
}